// ARPointerActor_72919954751906
// MI455X (gfx1250) — compile-verified
//
#include <hip/hip_runtime.h>
#include <math.h>

// ---------------- CDNA5 WMMA types ----------------
typedef __attribute__((ext_vector_type(2))) float v2f;
typedef __attribute__((ext_vector_type(8))) float v8f;

#define BSZ   16
#define NNODE 512
#define E     128
#define DI    256
#define TWOE  256
#define H     256
#define S     16
#define R     8

static __device__ __forceinline__ float sigmf(float x) { return 1.f / (1.f + expf(-x)); }

// D = A(16xK) * B(Kx16) + acc, f32 WMMA, K multiple of 4.
// A: LDS/global row-major, lda stride, rows indexed by lo.
// Bcol: pointer already offset to this tile's column (col = lo), ldb stride.
static __device__ __forceinline__ v8f wmma_tile_f32(const float* A, int lda,
                                                    const float* Bcol, int ldb,
                                                    int K, v8f acc, int lo, int hi) {
  for (int k = 0; k < K; k += 4) {
    v2f av, bv;
    av.x = A[lo * lda + k + 2 * hi];
    av.y = A[lo * lda + k + 2 * hi + 1];
    bv.x = Bcol[(size_t)(k + 2 * hi) * ldb];
    bv.y = Bcol[(size_t)(k + 2 * hi + 1) * ldb];
    acc = __builtin_amdgcn_wmma_f32_16x16x4_f32(false, av, false, bv, (short)0, acc,
                                                false, false);
  }
  return acc;
}

// ---------------- h = LN(batch @ embed_w) ----------------
__global__ void k_embed_ln(const float* __restrict__ batch, const float* __restrict__ W,
                           const float* __restrict__ g, const float* __restrict__ b,
                           float* __restrict__ h) {
  int r = blockIdx.x, e = threadIdx.x;           // r in [0,8192), e in [0,128)
  float x0 = batch[r * 2 + 0], x1 = batch[r * 2 + 1];
  float v = x0 * W[e] + x1 * W[E + e];
  __shared__ float red[E];
  red[e] = v; __syncthreads();
  for (int o = 64; o > 0; o >>= 1) { if (e < o) red[e] += red[e + o]; __syncthreads(); }
  float m = red[0] * (1.f / E); __syncthreads();
  float xm = v - m;
  red[e] = xm * xm; __syncthreads();
  for (int o = 64; o > 0; o >>= 1) { if (e < o) red[e] += red[e + o]; __syncthreads(); }
  float var = red[0] * (1.f / E);
  h[r * E + e] = xm * rsqrtf(var + 1e-5f) * g[e] + b[e];
}

// ---------------- generic f32 WMMA GEMM: C(MxN) = A(MxK) @ B(KxN) ----------------
// flipA: A sequence-row index n -> 511-n within each block of 512 rows.
__global__ void k_gemm_wmma(const float* __restrict__ A, const float* __restrict__ B,
                            float* __restrict__ C, int M, int N, int K,
                            int lda, int ldb, int ldc, int flipA) {
  int wid  = (blockIdx.x * blockDim.x + threadIdx.x) >> 5;
  int lane = threadIdx.x & 31;
  int lo = lane & 15, hi = lane >> 4;
  int tilesN = N >> 4;
  int tm = wid / tilesN, tn = wid - tm * tilesN;
  if (tm * 16 >= M) return;
  int mrow = tm * 16 + lo;
  if (flipA) mrow = (mrow & ~(NNODE - 1)) | ((NNODE - 1) - (mrow & (NNODE - 1)));
  const float* Arow = A + (size_t)mrow * lda;
  const float* Bcol = B + tn * 16 + lo;
  v8f acc = {0.f, 0.f, 0.f, 0.f, 0.f, 0.f, 0.f, 0.f};
  for (int k = 0; k < K; k += 4) {
    v2f av, bv;
    av.x = Arow[k + 2 * hi];
    av.y = Arow[k + 2 * hi + 1];
    bv.x = Bcol[(size_t)(k + 2 * hi) * ldb];
    bv.y = Bcol[(size_t)(k + 2 * hi + 1) * ldb];
    acc = __builtin_amdgcn_wmma_f32_16x16x4_f32(false, av, false, bv, (short)0, acc,
                                                false, false);
  }
#pragma unroll
  for (int rr = 0; rr < 8; rr++) {
    int m = tm * 16 + rr + 8 * hi;
    C[(size_t)m * ldc + tn * 16 + lo] = acc[rr];
  }
}

// ---------------- causal depthwise conv (K=4) + SiLU ----------------
__global__ void k_conv_silu(const float* __restrict__ xz, const float* __restrict__ cw,
                            const float* __restrict__ cb, float* __restrict__ xc) {
  int r = blockIdx.x, c = threadIdx.x;           // r in [0,8192), c in [0,256)
  int n = r & (NNODE - 1), base = r - n;
  float acc = cb[c];
#pragma unroll
  for (int k = 0; k < 4; k++) {
    int nn = n + k - 3;
    if (nn >= 0) acc += xz[(size_t)(base + nn) * (2 * DI) + c] * cw[c * 4 + k];
  }
  xc[(size_t)r * DI + c] = acc * sigmf(acc);
}

// ---------------- p = xc @ Wx  (256 -> 40) ----------------
__global__ void k_xproj(const float* __restrict__ xc, const float* __restrict__ Wx,
                        float* __restrict__ p) {
  int r = blockIdx.x, j = threadIdx.x;
  if (j >= R + 2 * S) return;
  const float* xr = xc + (size_t)r * DI;
  float acc = 0.f;
  for (int c = 0; c < DI; c++) acc += xr[c] * Wx[c * (R + 2 * S) + j];
  p[(size_t)r * (R + 2 * S) + j] = acc;
}

// ---------------- dt = softplus(p[:,:8] @ Wdt + bdt) ----------------
__global__ void k_dt(const float* __restrict__ p, const float* __restrict__ Wdt,
                     const float* __restrict__ bdt, float* __restrict__ dt) {
  int r = blockIdx.x, c = threadIdx.x;
  const float* pr = p + (size_t)r * (R + 2 * S);
  float acc = bdt[c];
#pragma unroll
  for (int i = 0; i < R; i++) acc += pr[i] * Wdt[i * DI + c];
  dt[(size_t)r * DI + c] = fmaxf(acc, 0.f) + log1pf(expf(-fabsf(acc)));
}

// ---------------- selective scan + D skip + SiLU(z) gate ----------------
// thread t: s = t&15, dch = (t>>4)&255, b = t>>12. S lanes in low lane bits -> shfl reduce.
__global__ void k_scan(const float* __restrict__ dt, const float* __restrict__ p,
                       const float* __restrict__ xc, const float* __restrict__ xz,
                       const float* __restrict__ Alog, const float* __restrict__ Dp,
                       float* __restrict__ yg) {
  int t = blockIdx.x * blockDim.x + threadIdx.x;   // 65536 threads
  int s = t & (S - 1);
  int grp = t >> 4;
  int dch = grp & (DI - 1);
  int b = grp >> 8;
  float A  = -expf(Alog[dch * S + s]);
  float Dv = Dp[dch];
  float hst = 0.f;
  for (int n = 0; n < NNODE; n++) {
    size_t r = (size_t)b * NNODE + n;
    float dtv = dt[r * DI + dch];
    float xcv = xc[r * DI + dch];
    float Bv  = p[r * (R + 2 * S) + R + s];
    float Cv  = p[r * (R + 2 * S) + R + S + s];
    hst = expf(dtv * A) * hst + dtv * Bv * xcv;
    float contrib = hst * Cv;
    contrib += __shfl_xor(contrib, 1, 16);
    contrib += __shfl_xor(contrib, 2, 16);
    contrib += __shfl_xor(contrib, 4, 16);
    contrib += __shfl_xor(contrib, 8, 16);
    if (s == 0) {
      float z = xz[r * (2 * DI) + DI + dch];
      yg[r * DI + dch] = (contrib + xcv * Dv) * (z * sigmf(z));
    }
  }
}

// ---------------- residual combine (+ flip of backward dir) ----------------
__global__ void k_combine(const float* __restrict__ h, const float* __restrict__ y0,
                          const float* __restrict__ y1, float* __restrict__ hout,
                          float* __restrict__ enc, int last) {
  int r = blockIdx.x, e = threadIdx.x;             // e in [0,128)
  int n = r & (NNODE - 1);
  int fr = (r - n) + (NNODE - 1 - n);
  float hv = h[(size_t)r * E + e];
  float a  = hv + y0[(size_t)r * E + e];
  float bb = hv + y1[(size_t)fr * E + e];
  if (last) {
    enc[(size_t)r * TWOE + e]     = a;
    enc[(size_t)r * TWOE + E + e] = bb;
  } else {
    hout[(size_t)r * E + e] = a + bb;
  }
}

// ---------------- LayerNorm over 256 dims, in place ----------------
__global__ void k_ln256(float* __restrict__ x, const float* __restrict__ g,
                        const float* __restrict__ b) {
  int r = blockIdx.x, c = threadIdx.x;
  __shared__ float red[TWOE];
  float v = x[(size_t)r * TWOE + c];
  red[c] = v; __syncthreads();
  for (int o = 128; o > 0; o >>= 1) { if (c < o) red[c] += red[c + o]; __syncthreads(); }
  float m = red[0] * (1.f / TWOE); __syncthreads();
  float xm = v - m;
  red[c] = xm * xm; __syncthreads();
  for (int o = 128; o > 0; o >>= 1) { if (c < o) red[c] += red[c + o]; __syncthreads(); }
  float var = red[0] * (1.f / TWOE);
  x[(size_t)r * TWOE + c] = xm * rsqrtf(var + 1e-5f) * g[c] + b[c];
}

// ---------------- graph = mean_n(enc) ----------------
__global__ void k_graph(const float* __restrict__ enc, float* __restrict__ graph) {
  int b = blockIdx.x, c = threadIdx.x;
  float acc = 0.f;
  for (int n = 0; n < NNODE; n++) acc += enc[((size_t)b * NNODE + n) * TWOE + c];
  graph[b * TWOE + c] = acc * (1.f / NNODE);
}

// ---------------- hdec = tanh(graph @ Wh0 + bh0) ----------------
__global__ void k_hdec(const float* __restrict__ graph, const float* __restrict__ Wh0,
                       const float* __restrict__ bh0, float* __restrict__ hdec) {
  int b = blockIdx.x, c = threadIdx.x;
  float acc = bh0[c];
  for (int i = 0; i < TWOE; i++) acc += graph[b * TWOE + i] * Wh0[i * H + c];
  hdec[b * H + c] = tanhf(acc);
}

// ---------------- persistent pointer decoder: 512 sequential steps ----------------
// One workgroup, 1024 threads = 32 waves. Batch dim (16) == WMMA M.
__global__ void __launch_bounds__(1024) k_decoder(
    const float* __restrict__ enc, const float* __restrict__ keys,
    const float* __restrict__ hdec, const float* __restrict__ Wq,
    const float* __restrict__ Wx, const float* __restrict__ Wh,
    const float* __restrict__ gb, float* __restrict__ out) {
  const int tid  = threadIdx.x;
  const int lane = tid & 31, wave = tid >> 5;
  const int lo = lane & 15, hi = lane >> 4;

  __shared__ float s_h[BSZ * H];        // 16 KB  hidden state
  __shared__ float s_q[BSZ * TWOE];     // 16 KB  query
  __shared__ float s_sel[BSZ * TWOE];   // 16 KB  selected enc row
  __shared__ float s_rz[BSZ * 2 * H];   // 32 KB  (xr+hr | xz+hz)
  __shared__ float s_xn[BSZ * H];       // 16 KB
  __shared__ float s_hn[BSZ * H];       // 16 KB
  __shared__ float s_red[1024];
  __shared__ int   s_redi[1024];
  __shared__ unsigned s_mask[BSZ * 16]; // 512-bit mask per batch
  __shared__ int s_nxt[BSZ];

  for (int i = tid; i < BSZ * H; i += 1024) s_h[i] = hdec[i];
  for (int i = tid; i < BSZ * 16; i += 1024) s_mask[i] = 0u;
  __syncthreads();

  float* out_perm  = out;
  float* out_probs = out + (size_t)BSZ * NNODE * NNODE;
  const float scale = 0.0625f;  // 1/sqrt(256)

  const int bq = tid >> 6;      // batch for logit phase (64 threads per batch)
  const int j  = tid & 63;

  for (int t = 0; t < NNODE; t++) {
    // ---- q = s_h @ Wq : waves 0..15 each compute one 16x16 N-tile ----
    if (wave < 16) {
      v8f acc = {0.f, 0.f, 0.f, 0.f, 0.f, 0.f, 0.f, 0.f};
      acc = wmma_tile_f32(s_h, H, Wq + wave * 16 + lo, TWOE, H, acc, lo, hi);
#pragma unroll
      for (int rr = 0; rr < 8; rr++) s_q[(rr + 8 * hi) * TWOE + wave * 16 + lo] = acc[rr];
    }
    __syncthreads();

    // ---- logits[b,n] = scale * q[b]·keys[b,n]; masked -> -1e9 ----
    float lv[8];
    float bestv = -3.0e38f; int besti = 1 << 30;
    const float4* qv4 = (const float4*)(s_q + bq * TWOE);
    const unsigned* mb = s_mask + bq * 16;
#pragma unroll
    for (int i = 0; i < 8; i++) {
      int n = j + (i << 6);
      const float4* kv4 = (const float4*)(keys + ((size_t)(bq * NNODE + n)) * TWOE);
      float acc = 0.f;
      for (int c = 0; c < TWOE / 4; c++) {
        float4 kv = kv4[c], qv = qv4[c];
        acc += kv.x * qv.x + kv.y * qv.y + kv.z * qv.z + kv.w * qv.w;
      }
      float l = ((mb[n >> 5] >> (n & 31)) & 1u) ? -1.0e9f : acc * scale;
      lv[i] = l;
      if (l > bestv || (l == bestv && n < besti)) { bestv = l; besti = n; }
    }
    // max + argmax over the 64 threads of this batch (first-occurrence ties)
    s_red[tid] = bestv; s_redi[tid] = besti; __syncthreads();
    for (int o = 32; o > 0; o >>= 1) {
      if (j < o) {
        float ov = s_red[tid + o]; int oi = s_redi[tid + o];
        if (ov > s_red[tid] || (ov == s_red[tid] && oi < s_redi[tid])) {
          s_red[tid] = ov; s_redi[tid] = oi;
        }
      }
      __syncthreads();
    }
    float m = s_red[bq << 6];
    int nxt  = s_redi[bq << 6];
    if (j == 0) s_nxt[bq] = nxt;
    __syncthreads();

    // ---- softmax ----
    float pv[8], psum = 0.f;
#pragma unroll
    for (int i = 0; i < 8; i++) { pv[i] = expf(lv[i] - m); psum += pv[i]; }
    s_red[tid] = psum; __syncthreads();
    for (int o = 32; o > 0; o >>= 1) { if (j < o) s_red[tid] += s_red[tid + o]; __syncthreads(); }
    float inv = 1.f / s_red[bq << 6];

    size_t obase = ((size_t)bq * NNODE + t) * NNODE;
#pragma unroll
    for (int i = 0; i < 8; i++) {
      int n = j + (i << 6);
      out_probs[obase + n] = pv[i] * inv;
      out_perm[obase + n]  = (n == nxt) ? 1.f : 0.f;
    }
    if (j == 0) s_mask[bq * 16 + (nxt >> 5)] |= (1u << (nxt & 31));
    __syncthreads();

    // ---- gather selected rows ----
    for (int i = tid; i < BSZ * TWOE; i += 1024) {
      int bb = i >> 8, c = i & (TWOE - 1);
      s_sel[i] = enc[((size_t)(bb * NNODE + s_nxt[bb])) * TWOE + c];
    }
    __syncthreads();

    // ---- GRU GEMMs via WMMA: 64 tiles over 32 waves ----
    for (int tix = wave; tix < 64; tix += 32) {
      v8f acc = {0.f, 0.f, 0.f, 0.f, 0.f, 0.f, 0.f, 0.f};
      if (tix < 32) {               // r/z halves: sel@Wx[:, :512] + h@Wh[:, :512]
        int col = tix * 16 + lo;
        acc = wmma_tile_f32(s_sel, TWOE, Wx + col, 3 * H, TWOE, acc, lo, hi);
        acc = wmma_tile_f32(s_h,   H,    Wh + col, 3 * H, H,    acc, lo, hi);
#pragma unroll
        for (int rr = 0; rr < 8; rr++) s_rz[(rr + 8 * hi) * (2 * H) + col] = acc[rr];
      } else if (tix < 48) {        // xn = sel @ Wx[:, 512:768]
        int col = 2 * H + (tix - 32) * 16 + lo;
        acc = wmma_tile_f32(s_sel, TWOE, Wx + col, 3 * H, TWOE, acc, lo, hi);
#pragma unroll
        for (int rr = 0; rr < 8; rr++)
          s_xn[(rr + 8 * hi) * H + (tix - 32) * 16 + lo] = acc[rr];
      } else {                      // hn = h @ Wh[:, 512:768]
        int col = 2 * H + (tix - 48) * 16 + lo;
        acc = wmma_tile_f32(s_h, H, Wh + col, 3 * H, H, acc, lo, hi);
#pragma unroll
        for (int rr = 0; rr < 8; rr++)
          s_hn[(rr + 8 * hi) * H + (tix - 48) * 16 + lo] = acc[rr];
      }
    }
    __syncthreads();

    // ---- GRU elementwise update (in place on s_h) ----
    for (int i = tid; i < BSZ * H; i += 1024) {
      int bb = i >> 8, c = i & (H - 1);
      float rg = sigmf(s_rz[bb * 2 * H + c] + gb[c]);
      float zg = sigmf(s_rz[bb * 2 * H + H + c] + gb[H + c]);
      float ng = tanhf(s_xn[bb * H + c] + gb[2 * H + c] + rg * s_hn[bb * H + c]);
      float hv = s_h[bb * H + c];
      s_h[bb * H + c] = (1.f - zg) * ng + zg * hv;
    }
    __syncthreads();
  }
}

// ---------------- host launcher ----------------
extern "C" void kernel_launch(void* const* d_in, const int* in_sizes, int n_in,
                              void* d_out, int out_size, void* d_ws, size_t ws_size,
                              hipStream_t stream) {
  (void)in_sizes; (void)n_in; (void)out_size; (void)ws_size;
  const float* batch   = (const float*)d_in[0];
  const float* embed_w = (const float*)d_in[1];
  const float* ln1_g   = (const float*)d_in[2];
  const float* ln1_b   = (const float*)d_in[3];
  const float* in_proj = (const float*)d_in[4];
  const float* conv_w  = (const float*)d_in[5];
  const float* conv_b  = (const float*)d_in[6];
  const float* x_proj  = (const float*)d_in[7];
  const float* dt_w    = (const float*)d_in[8];
  const float* dt_b    = (const float*)d_in[9];
  const float* A_log   = (const float*)d_in[10];
  const float* Dp      = (const float*)d_in[11];
  const float* out_prj = (const float*)d_in[12];
  const float* ln2_g   = (const float*)d_in[13];
  const float* ln2_b   = (const float*)d_in[14];
  const float* Wk      = (const float*)d_in[15];
  const float* Wh0     = (const float*)d_in[16];
  const float* bh0     = (const float*)d_in[17];
  const float* Wq      = (const float*)d_in[18];
  const float* gru_Wx  = (const float*)d_in[19];
  const float* gru_Wh  = (const float*)d_in[20];
  const float* gru_b   = (const float*)d_in[21];

  const int ROWS = BSZ * NNODE;  // 8192
  float* ws   = (float*)d_ws;    // needs ~73 MB
  float* h    = ws;                       // 8192*128
  float* xz   = h    + (size_t)ROWS * E;          // 8192*512
  float* xc   = xz   + (size_t)ROWS * 2 * DI;     // 8192*256
  float* p    = xc   + (size_t)ROWS * DI;         // 8192*40
  float* dtb  = p    + (size_t)ROWS * (R + 2 * S);// 8192*256
  float* yg   = dtb  + (size_t)ROWS * DI;         // 8192*256
  float* yo0  = yg   + (size_t)ROWS * DI;         // 8192*128
  float* yo1  = yo0  + (size_t)ROWS * E;          // 8192*128
  float* enc  = yo1  + (size_t)ROWS * E;          // 8192*256
  float* keys = enc  + (size_t)ROWS * TWOE;       // 8192*256
  float* grph = keys + (size_t)ROWS * TWOE;       // 16*256
  float* hdec = grph + (size_t)BSZ * TWOE;        // 16*256

  k_embed_ln<<<ROWS, E, 0, stream>>>(batch, embed_w, ln1_g, ln1_b, h);

  for (int l = 0; l < 2; l++) {
    for (int d = 0; d < 2; d++) {
      int od = l * 2 + d;
      // xz = x @ Wi  (8192 x 512, K=128), flip rows for backward dir
      k_gemm_wmma<<<2048, 256, 0, stream>>>(h, in_proj + (size_t)od * E * 2 * DI, xz,
                                            ROWS, 2 * DI, E, E, 2 * DI, 2 * DI, d);
      k_conv_silu<<<ROWS, DI, 0, stream>>>(xz, conv_w + od * DI * 4, conv_b + od * DI, xc);
      k_xproj<<<ROWS, 64, 0, stream>>>(xc, x_proj + od * DI * (R + 2 * S), p);
      k_dt<<<ROWS, DI, 0, stream>>>(p, dt_w + od * R * DI, dt_b + od * DI, dtb);
      k_scan<<<256, 256, 0, stream>>>(dtb, p, xc, xz, A_log + od * DI * S, Dp + od * DI, yg);
      // yo = yg @ Wo  (8192 x 128, K=256)
      k_gemm_wmma<<<512, 256, 0, stream>>>(yg, out_prj + (size_t)od * DI * E,
                                           d ? yo1 : yo0, ROWS, E, DI, DI, E, E, 0);
    }
    k_combine<<<ROWS, E, 0, stream>>>(h, yo0, yo1, h, enc, l == 1);
  }

  k_ln256<<<ROWS, TWOE, 0, stream>>>(enc, ln2_g, ln2_b);
  k_graph<<<BSZ, TWOE, 0, stream>>>(enc, grph);
  // keys = enc @ Wk  (8192 x 256, K=256)
  k_gemm_wmma<<<1024, 256, 0, stream>>>(enc, Wk, keys, ROWS, TWOE, TWOE, TWOE, TWOE, TWOE, 0);
  k_hdec<<<BSZ, H, 0, stream>>>(grph, Wh0, bh0, hdec);

  k_decoder<<<1, 1024, 0, stream>>>(enc, keys, hdec, Wq, gru_Wx, gru_Wh, gru_b,
                                    (float*)d_out);
}